// Shift_tcn_student_32186484916745
// MI455X (gfx1250) — compile-verified
//
#include <hip/hip_runtime.h>
#include <hip/hip_bf16.h>

typedef __attribute__((ext_vector_type(2))) float v2f;
typedef __attribute__((ext_vector_type(8))) float v8f;

// Problem constants
#define NB   64
#define CC   128
#define TT   300
#define VV   25
#define TVN  7500     // T*V
#define DD   256
#define RR   32
#define NTV  480000   // N*T*V

// GEMM tiling
#define MB      64
#define DB      128
#define KC      128
#define A_STR   132   // 128 + 4 pad: conflict-free A frag b64 reads
#define BP_STR  288   // k-pair row stride (DB*2 + 32 pad): half-waves on disjoint bank halves
#define C_STR   132
#define SMEM_FLOATS (MB * A_STR + (KC / 2) * BP_STR)  // 8448 + 18432 = 26880 floats
#define SMEM_BYTES  (SMEM_FLOATS * 4)                 // 107520 bytes dynamic LDS

// CDNA5 async global->LDS path (guarded: falls back to plain staging if absent)
#if defined(__has_builtin)
#if __has_builtin(__builtin_amdgcn_global_load_async_to_lds_b32)
#define HAVE_ASYNC_LDS 1
#endif
#if __has_builtin(__builtin_amdgcn_s_wait_asynccnt)
#define WAIT_ASYNC() __builtin_amdgcn_s_wait_asynccnt(0)
#else
#define WAIT_ASYNC() asm volatile("s_wait_asynccnt 0x0" ::: "memory")
#endif
#endif
#ifndef HAVE_ASYNC_LDS
#define HAVE_ASYNC_LDS 0
#endif

#define AS1 __attribute__((address_space(1)))
#define AS3 __attribute__((address_space(3)))

#if HAVE_ASYNC_LDS
__device__ __forceinline__ AS1 int* gcast(const float* p) {
  float* q = const_cast<float*>(p);
  return (AS1 int*)q;     // generic -> global addrspacecast + bitcast
}
__device__ __forceinline__ AS3 int* lcast(float* p) {
  return (AS3 int*)p;     // generic -> LDS addrspacecast + bitcast
}
#endif

// ---------------- helpers ----------------
__device__ __forceinline__ float block_reduce_sum(float v, float* sbuf) {
  int tid = threadIdx.x;
  sbuf[tid] = v;
  __syncthreads();
  for (int s = blockDim.x >> 1; s > 0; s >>= 1) {
    if (tid < s) sbuf[tid] += sbuf[tid + s];
    __syncthreads();
  }
  float r = sbuf[0];
  __syncthreads();
  return r;
}

__device__ __forceinline__ void shift_coef(float sh, int t, int& i0c, int& i1c,
                                           float& w0, float& w1) {
  float pos = (float)t + sh;
  float f = floorf(pos);
  int i0 = (int)f;
  float frac = pos - f;
  int i1 = i0 + 1;
  w0 = (i0 >= 0 && i0 < TT) ? (1.0f - frac) : 0.0f;
  w1 = (i1 >= 0 && i1 < TT) ? frac : 0.0f;
  i0c = i0 < 0 ? 0 : (i0 > TT - 1 ? TT - 1 : i0);
  i1c = i1 < 0 ? 0 : (i1 > TT - 1 ? TT - 1 : i1);
}

// ---------------- 1: BN1 stats -> scale/bias ----------------
__global__ __launch_bounds__(256) void k_bn1_stats(const float* __restrict__ x,
                                                   const float* __restrict__ gamma,
                                                   const float* __restrict__ beta,
                                                   float* __restrict__ scale1,
                                                   float* __restrict__ bias1) {
  __shared__ float red[256];
  int c = blockIdx.x;
  float s = 0.f, ss = 0.f;
  for (int i = threadIdx.x; i < NTV; i += 256) {
    int n = i / TVN, m = i - n * TVN;
    float v = x[(size_t)n * CC * TVN + (size_t)c * TVN + m];
    s += v; ss += v * v;
  }
  float ts = block_reduce_sum(s, red);
  float tss = block_reduce_sum(ss, red);
  if (threadIdx.x == 0) {
    float mean = ts / (float)NTV;
    float var = tss / (float)NTV - mean * mean;
    float inv = rsqrtf(var + 1e-5f);
    float sc = gamma[c] * inv;
    scale1[c] = sc;
    bias1[c] = beta[c] - mean * sc;
  }
}

// ---------------- 2: BN1 + temporal_shift_in + pooled mean ----------------
__global__ __launch_bounds__(256) void k_bnshift_in(const float* __restrict__ x,
                                                    const float* __restrict__ scale1,
                                                    const float* __restrict__ bias1,
                                                    const float* __restrict__ shift_in,
                                                    float* __restrict__ y,
                                                    float* __restrict__ p) {
  __shared__ float red[256];
  int c = blockIdx.x, n = blockIdx.y;
  float sc = scale1[c], bi = bias1[c], sh = shift_in[c];
  size_t base = ((size_t)n * CC + c) * TVN;
  float sum = 0.f;
  for (int m = threadIdx.x; m < TVN; m += 256) {
    int t = m / VV, v = m - t * VV;
    int i0c, i1c; float w0, w1;
    shift_coef(sh, t, i0c, i1c, w0, w1);
    float x0 = x[base + (size_t)i0c * VV + v];
    float x1 = x[base + (size_t)i1c * VV + v];
    float val = w0 * (sc * x0 + bi) + w1 * (sc * x1 + bi);
    y[base + m] = val;
    sum += val;
  }
  float tot = block_reduce_sum(sum, red);
  if (threadIdx.x == 0) p[n * CC + c] = tot / (float)TVN;
}

// ---------------- 3: gating MLP + softmax ----------------
__global__ __launch_bounds__(64) void k_gate(const float* __restrict__ p,
                                             const float* __restrict__ fc1_w,
                                             const float* __restrict__ fc1_b,
                                             const float* __restrict__ fc2_w,
                                             const float* __restrict__ fc2_b,
                                             const int* __restrict__ epoch_p,
                                             float* __restrict__ gate) {
  int n = threadIdx.x;
  if (n >= NB) return;
  float h[RR];
  for (int r = 0; r < RR; ++r) {
    float acc = fc1_b[r];
    for (int c = 0; c < CC; ++c) acc += p[n * CC + c] * fc1_w[r * CC + c];
    h[r] = fmaxf(acc, 0.0f);
  }
  float lg[4];
  for (int k = 0; k < 4; ++k) {
    float acc = fc2_b[k];
    for (int r = 0; r < RR; ++r) acc += h[r] * fc2_w[k * RR + r];
    lg[k] = acc;
  }
  int epoch = epoch_p[0];
  float tao = (epoch < 60) ? (-(29.0f / 60.0f) * (float)epoch + 30.0f) : 1.0f;
  float mx = lg[0];
  for (int k = 1; k < 4; ++k) mx = fmaxf(mx, lg[k]);
  float e[4], sum = 0.f;
  for (int k = 0; k < 4; ++k) { e[k] = __expf((lg[k] - mx) / tao); sum += e[k]; }
  for (int k = 0; k < 4; ++k) gate[n * 4 + k] = e[k] / sum;
}

// ---------------- 4: mix expert weights per sample ----------------
__global__ __launch_bounds__(256) void k_mix(const float* __restrict__ tw,
                                             const float* __restrict__ tb,
                                             const float* __restrict__ gate,
                                             float* __restrict__ Wn,
                                             float* __restrict__ bvec) {
  int n = blockIdx.x;
  float g0 = gate[n * 4 + 0], g1 = gate[n * 4 + 1];
  float g2 = gate[n * 4 + 2], g3 = gate[n * 4 + 3];
  const int CD = CC * DD;
  for (int i = threadIdx.x; i < CD; i += 256) {
    Wn[(size_t)n * CD + i] = g0 * tw[i] + g1 * tw[CD + i] +
                             g2 * tw[2 * CD + i] + g3 * tw[3 * CD + i];
  }
  for (int i = threadIdx.x; i < DD; i += 256) {
    bvec[n * DD + i] = g0 * tb[i] + g1 * tb[DD + i] +
                       g2 * tb[2 * DD + i] + g3 * tb[3 * DD + i];
  }
}

// ---------------- 5: WMMA GEMM  z = relu(y^T W + b) ----------------
__global__ __launch_bounds__(256) void k_gemm(const float* __restrict__ y,
                                              const float* __restrict__ Wn,
                                              const float* __restrict__ bvec,
                                              float* __restrict__ z) {
  extern __shared__ float smem[];
  float* As = smem;                 // [MB][A_STR]              (m-major)
  float* Bs = smem + MB * A_STR;    // [KC/2][BP_STR] k-pair interleaved: (c,d) -> [(c>>1)][d*2+(c&1)]
  float* Cs = smem;                 // reused post-barrier: [MB][C_STR]

  int mb = blockIdx.x * MB;
  int db = blockIdx.y * DB;
  int n  = blockIdx.z;
  const float* yrow = y  + (size_t)n * CC * TVN;
  const float* wmat = Wn + (size_t)n * CC * DD;
  int tid = threadIdx.x;

  // Stage A: 64 rows (m) x 128 cols (c), coalesced over m (transpose into LDS)
  for (int i = tid; i < MB * KC; i += 256) {
    int c = i >> 6, m = i & 63;
    int mg = mb + m; if (mg > TVN - 1) mg = TVN - 1;
#if HAVE_ASYNC_LDS
    __builtin_amdgcn_global_load_async_to_lds_b32(
        gcast(yrow + (size_t)c * TVN + mg),
        lcast(&As[m * A_STR + c]), 0, 0);
#else
    As[m * A_STR + c] = yrow[(size_t)c * TVN + mg];
#endif
  }
  // Stage B: 128 rows (c) x 128 cols (d), coalesced over d, k-pair interleave
  for (int i = tid; i < KC * DB; i += 256) {
    int c = i >> 7, d = i & 127;
#if HAVE_ASYNC_LDS
    __builtin_amdgcn_global_load_async_to_lds_b32(
        gcast(wmat + (size_t)c * DD + db + d),
        lcast(&Bs[(c >> 1) * BP_STR + d * 2 + (c & 1)]), 0, 0);
#else
    Bs[(c >> 1) * BP_STR + d * 2 + (c & 1)] = wmat[(size_t)c * DD + db + d];
#endif
  }
#if HAVE_ASYNC_LDS
  WAIT_ASYNC();
#endif
  __syncthreads();

  int lane = tid & 31;
  int wave = tid >> 5;
  int mt = wave & 3;                 // m tile within block (0..3)
  int ng = wave >> 2;                // d group (0..1), 64 cols each
  int row = mt * 16 + (lane & 15);
  int hk  = (lane >> 4) << 1;        // 0 or 2 (K pair select per half-wave)
  int colb = ng * 64 + (lane & 15);

  v8f acc0 = {}, acc1 = {}, acc2 = {}, acc3 = {};
  // A frag: lane reads A[row][k+hk .. k+hk+1] as one aligned b64
  const float* arow = &As[row * A_STR + hk];
  // B frag: lane reads pair row kp = k/2 + (lane>>4), cols interleaved -> one aligned b64
  const float* brow = &Bs[(lane >> 4) * BP_STR + colb * 2];

#pragma unroll
  for (int k = 0; k < KC; k += 4) {
    v2f a = *(const v2f*)(arow + k);
    const float* bk = brow + (k >> 1) * BP_STR;
    v2f bA = *(const v2f*)(bk +  0);   // cols +0
    v2f bB = *(const v2f*)(bk + 32);   // cols +16
    v2f bC = *(const v2f*)(bk + 64);   // cols +32
    v2f bD = *(const v2f*)(bk + 96);   // cols +48
    acc0 = __builtin_amdgcn_wmma_f32_16x16x4_f32(false, a, false, bA, (short)0, acc0, false, false);
    acc1 = __builtin_amdgcn_wmma_f32_16x16x4_f32(false, a, false, bB, (short)0, acc1, false, false);
    acc2 = __builtin_amdgcn_wmma_f32_16x16x4_f32(false, a, false, bC, (short)0, acc2, false, false);
    acc3 = __builtin_amdgcn_wmma_f32_16x16x4_f32(false, a, false, bD, (short)0, acc3, false, false);
  }
  __syncthreads();   // everyone done reading As/Bs before Cs overwrite

  // C/D layout: VGPR r -> M = r + 8*(lane>=16), N = lane&15
#pragma unroll
  for (int r = 0; r < 8; ++r) {
    int mrow = mt * 16 + r + ((lane >> 4) << 3);
    int cbase = mrow * C_STR + ng * 64 + (lane & 15);
    Cs[cbase +  0] = acc0[r];
    Cs[cbase + 16] = acc1[r];
    Cs[cbase + 32] = acc2[r];
    Cs[cbase + 48] = acc3[r];
  }
  __syncthreads();

  // Coalesced store: contiguous m segments per d; + bias, ReLU
  for (int i = tid; i < MB * DB; i += 256) {
    int d = i >> 6, m = i & 63;
    int mg = mb + m;
    if (mg < TVN) {
      float bias = bvec[n * DD + db + d];
      float val = fmaxf(Cs[m * C_STR + d] + bias, 0.0f);
      z[(size_t)n * DD * TVN + (size_t)(db + d) * TVN + mg] = val;
    }
  }
}

// ---------------- 6a: zero BN2 accumulators ----------------
__global__ void k_zero(float* __restrict__ stat2) {
  int i = blockIdx.x * blockDim.x + threadIdx.x;
  if (i < 2 * DD) stat2[i] = 0.0f;
}

// ---------------- 6: shift_out stats (sum/sumsq per d) ----------------
__global__ __launch_bounds__(256) void k_stats2(const float* __restrict__ z,
                                                const float* __restrict__ shift_out,
                                                float* __restrict__ stat2) {
  __shared__ float row[TVN];
  __shared__ float red[256];
  int d = blockIdx.x, n = blockIdx.y;
  const float* zr = z + ((size_t)n * DD + d) * TVN;
  for (int m = threadIdx.x; m < TVN; m += 256) row[m] = zr[m];
  __syncthreads();
  float sh = shift_out[d];
  float s = 0.f, ss = 0.f;
  for (int m = threadIdx.x; m < TVN; m += 256) {
    int t = m / VV, v = m - t * VV;
    int i0c, i1c; float w0, w1;
    shift_coef(sh, t, i0c, i1c, w0, w1);
    float val = w0 * row[i0c * VV + v] + w1 * row[i1c * VV + v];
    s += val; ss += val * val;
  }
  float ts = block_reduce_sum(s, red);
  float tss = block_reduce_sum(ss, red);
  if (threadIdx.x == 0) {
    atomicAdd(&stat2[d], ts);
    atomicAdd(&stat2[DD + d], tss);
  }
}

// ---------------- 6b: finalize BN2 scale/bias ----------------
__global__ void k_fin2(const float* __restrict__ stat2,
                       const float* __restrict__ gamma,
                       const float* __restrict__ beta,
                       float* __restrict__ scale2,
                       float* __restrict__ bias2) {
  int d = blockIdx.x * blockDim.x + threadIdx.x;
  if (d < DD) {
    float mean = stat2[d] / (float)NTV;
    float var = stat2[DD + d] / (float)NTV - mean * mean;
    float inv = rsqrtf(var + 1e-5f);
    float sc = gamma[d] * inv;
    scale2[d] = sc;
    bias2[d] = beta[d] - mean * sc;
  }
}

// ---------------- 7: in-place shift_out + BN2 ----------------
__global__ __launch_bounds__(256) void k_apply2(float* __restrict__ z,
                                                const float* __restrict__ shift_out,
                                                const float* __restrict__ scale2,
                                                const float* __restrict__ bias2) {
  __shared__ float row[TVN];
  int d = blockIdx.x, n = blockIdx.y;
  float* zr = z + ((size_t)n * DD + d) * TVN;
  for (int m = threadIdx.x; m < TVN; m += 256) row[m] = zr[m];
  __syncthreads();
  float sh = shift_out[d], sc = scale2[d], bi = bias2[d];
  for (int m = threadIdx.x; m < TVN; m += 256) {
    int t = m / VV, v = m - t * VV;
    int i0c, i1c; float w0, w1;
    shift_coef(sh, t, i0c, i1c, w0, w1);
    float val = w0 * row[i0c * VV + v] + w1 * row[i1c * VV + v];
    zr[m] = sc * val + bi;
  }
}

// ---------------- launch ----------------
extern "C" void kernel_launch(void* const* d_in, const int* in_sizes, int n_in,
                              void* d_out, int out_size, void* d_ws, size_t ws_size,
                              hipStream_t stream) {
  const float* x         = (const float*)d_in[0];
  const int*   epoch     = (const int*)d_in[1];
  const float* bn1_gamma = (const float*)d_in[2];
  const float* bn1_beta  = (const float*)d_in[3];
  const float* bn2_gamma = (const float*)d_in[4];
  const float* bn2_beta  = (const float*)d_in[5];
  const float* shift_in  = (const float*)d_in[6];
  const float* shift_out = (const float*)d_in[7];
  const float* fc1_w     = (const float*)d_in[8];
  const float* fc1_b     = (const float*)d_in[9];
  const float* fc2_w     = (const float*)d_in[10];
  const float* fc2_b     = (const float*)d_in[11];
  const float* tw        = (const float*)d_in[12];
  const float* tb        = (const float*)d_in[13];
  float* out = (float*)d_out;
  float* ws  = (float*)d_ws;

  // workspace layout (floats)
  float* scale1 = ws;                         // 128
  float* bias1  = scale1 + 128;               // 128
  float* p      = bias1 + 128;                // 8192
  float* gate   = p + NB * CC;                // 256
  float* bvec   = gate + NB * 4;              // 16384
  float* stat2  = bvec + NB * DD;             // 512
  float* scale2 = stat2 + 2 * DD;             // 256
  float* bias2  = scale2 + DD;                // 256
  float* Wn     = bias2 + DD;                 // 64*128*256
  float* y      = Wn + (size_t)NB * CC * DD;  // 64*128*7500

  k_bn1_stats<<<dim3(CC), 256, 0, stream>>>(x, bn1_gamma, bn1_beta, scale1, bias1);
  k_bnshift_in<<<dim3(CC, NB), 256, 0, stream>>>(x, scale1, bias1, shift_in, y, p);
  k_gate<<<1, 64, 0, stream>>>(p, fc1_w, fc1_b, fc2_w, fc2_b, epoch, gate);
  k_mix<<<dim3(NB), 256, 0, stream>>>(tw, tb, gate, Wn, bvec);

  int mtiles = (TVN + MB - 1) / MB;           // 118
  k_gemm<<<dim3(mtiles, DD / DB, NB), 256, SMEM_BYTES, stream>>>(y, Wn, bvec, out);

  k_zero<<<2, 256, 0, stream>>>(stat2);
  k_stats2<<<dim3(DD, NB), 256, 0, stream>>>(out, shift_out, stat2);
  k_fin2<<<1, 256, 0, stream>>>(stat2, bn2_gamma, bn2_beta, scale2, bias2);
  k_apply2<<<dim3(DD, NB), 256, 0, stream>>>(out, shift_out, scale2, bias2);
}